// NeuralPoissonPlain_7456063226615
// MI455X (gfx1250) — compile-verified
//
#include <hip/hip_runtime.h>
#include <hip/hip_bf16.h>
#include <stdint.h>

// ---------------------------------------------------------------------------
// Sparse block-grid trilinear interpolation + analytic gradient + validity
// mask for MI455X (gfx1250).
//
// gfx1250-specific paths used:
//   * TENSOR_LOAD_TO_LDS (TDM) DMAs the whole 128KB block table into LDS in a
//     single instruction per workgroup (wave 0 issues; EXEC ignored by TDM).
//   * s_wait_tensorcnt 0 to fence the DMA before the workgroup barrier.
//   * wave32 blocks (512 threads = 16 waves); ds_load gathers for the 8
//     block-table lookups per point.
// ---------------------------------------------------------------------------

typedef __attribute__((ext_vector_type(4))) unsigned int v4u;
typedef __attribute__((ext_vector_type(8))) int          v8i;
typedef __attribute__((ext_vector_type(4))) int          v4i;

#define GRID_RES      32
#define TABLE_N       (GRID_RES * GRID_RES * GRID_RES)   // 32768 int32
#define TABLE_BYTES   (TABLE_N * 4)                      // 131072 bytes
#define N_CELLS       256
#define EMB_PER_BLOCK 512                                // GRID_DIM^3

__global__ void np_sparse_interp_kernel(const float* __restrict__ pos,
                                        const float* __restrict__ emb,
                                        const int*   __restrict__ table,
                                        float*       __restrict__ out,
                                        int npts)
{
    extern __shared__ int s_table[];   // 128KB dynamic LDS, offset 0

    // ---- Stage block_table (128KB) into LDS via the Tensor Data Mover ----
    if (threadIdx.x < 32u) {           // wave 0 only issues the DMA
        const uint64_t ga = (uint64_t)(uintptr_t)table;

        v4u g0;
        g0[0] = 1u;                                   // count=1 (valid), user mode
        g0[1] = 0u;                                   // lds_addr = 0 (LDS base)
        g0[2] = (unsigned)(ga & 0xFFFFFFFFu);         // global_addr[31:0]
        g0[3] = ((unsigned)(ga >> 32) & 0x01FFFFFFu)  // global_addr[56:32]
              | (2u << 30);                           // type = 2 ("image")

        v8i g1;
        g1[0] = (int)0x00020000;        // workgroup_mask=0, data_size=2 (4B)
        g1[1] = (int)0x80000000u;       // tensor_dim0[15:0]=32768 in [31:16]
        g1[2] = (int)0x00010000;        // tensor_dim0 hi=0; tensor_dim1=1
        g1[3] = (int)0x80000000u;       // tensor_dim1 hi=0; tile_dim0=32768
        g1[4] = 1;                      // tile_dim1=1, tile_dim2=0
        g1[5] = TABLE_N;                // tensor_dim0_stride[31:0]=32768
        g1[6] = 0;                      // stride0 hi=0, stride1 lo=0
        g1[7] = 0;                      // stride1 hi=0

        v4i gz4 = {0, 0, 0, 0};         // groups 2/3: dims >1D unused
        v8i gz8 = {0, 0, 0, 0, 0, 0, 0, 0};

        __builtin_amdgcn_tensor_load_to_lds(g0, g1, gz4, gz4, gz8, 0);
        __builtin_amdgcn_s_wait_tensorcnt(0);
    }
    __syncthreads();                    // LDS table visible to all 16 waves

    const float scale = 0.5f * (float)(N_CELLS - 1);   // 127.5
    const int   stride = (int)(gridDim.x * blockDim.x);

    for (int i = (int)(blockIdx.x * blockDim.x + threadIdx.x); i < npts;
         i += stride) {
        const float px = pos[3 * i + 0];
        const float py = pos[3 * i + 1];
        const float pz = pos[3 * i + 2];

        // world -> cell coords
        const float xc = (px + 1.0f) * scale;
        const float yc = (py + 1.0f) * scale;
        const float zc = (pz + 1.0f) * scale;

        float fbx = fminf(fmaxf(floorf(xc), 0.0f), 254.0f);
        float fby = fminf(fmaxf(floorf(yc), 0.0f), 254.0f);
        float fbz = fminf(fmaxf(floorf(zc), 0.0f), 254.0f);

        const int bx = (int)fbx, by = (int)fby, bz = (int)fbz;
        const float fx = xc - fbx, fy = yc - fby, fz = zc - fbz;

        // per-axis corner weights: w[0]=1-f (d=0), w[1]=f (d=1)
        const float wx[2] = {1.0f - fx, fx};
        const float wy[2] = {1.0f - fy, fy};
        const float wz[2] = {1.0f - fz, fz};

        // block-table row offsets and local-cell offsets for both corners/axis
        const int tx[2] = {(bx >> 3) * 1024, ((bx + 1) >> 3) * 1024};
        const int ty[2] = {(by >> 3) * 32,   ((by + 1) >> 3) * 32};
        const int tz[2] = {(bz >> 3),        ((bz + 1) >> 3)};
        const int lx[2] = {(bx & 7) * 64,    ((bx + 1) & 7) * 64};
        const int ly[2] = {(by & 7) * 8,     ((by + 1) & 7) * 8};
        const int lz[2] = {(bz & 7),         ((bz + 1) & 7)};

        float acc = 0.0f, gx = 0.0f, gy = 0.0f, gzs = 0.0f;
        int vall = 1;

        #pragma unroll
        for (int c = 0; c < 8; ++c) {
            const int dx = (c >> 2) & 1;
            const int dy = (c >> 1) & 1;
            const int dz = c & 1;

            const int bidx  = s_table[tx[dx] + ty[dy] + tz[dz]];  // LDS gather
            const int valid = (bidx >= 0) ? 1 : 0;
            vall &= valid;

            const int li   = lx[dx] + ly[dy] + lz[dz];
            const int eidx = (bidx > 0 ? bidx : 0) * EMB_PER_BLOCK + li;
            const float e  = emb[eidx];                           // L2 gather
            const float es = valid ? e : 0.0f;                    // where(valid,...)

            const float sx = dx ? 1.0f : -1.0f;
            const float sy = dy ? 1.0f : -1.0f;
            const float sz = dz ? 1.0f : -1.0f;

            acc = fmaf(wx[dx] * wy[dy] * wz[dz], es, acc);
            gx  = fmaf(sx * wy[dy] * wz[dz],     es, gx);
            gy  = fmaf(sy * wx[dx] * wz[dz],     es, gy);
            gzs = fmaf(sz * wx[dx] * wy[dy],     es, gzs);
        }

        // chain rule: d(cell)/d(pos) = 0.5*(N_CELLS-1) = 127.5
        out[i]                   = acc;
        out[npts + 3 * i + 0]    = gx  * scale;
        out[npts + 3 * i + 1]    = gy  * scale;
        out[npts + 3 * i + 2]    = gzs * scale;
        out[4 * npts + i]        = vall ? 1.0f : 0.0f;
    }
}

extern "C" void kernel_launch(void* const* d_in, const int* in_sizes, int n_in,
                              void* d_out, int out_size, void* d_ws, size_t ws_size,
                              hipStream_t stream) {
    const float* positions  = (const float*)d_in[0];   // [N,3]
    const float* embeddings = (const float*)d_in[1];   // [12000,512,1]
    const int*   block_tbl  = (const int*)d_in[2];     // [32,32,32]
    float*       out        = (float*)d_out;           // N + 3N + N floats

    const int npts = in_sizes[0] / 3;

    // 512 threads = 16 wave32s; 128KB LDS -> two blocks co-resident per WGP
    // (256KB of the 320KB pool). Grid-stride keeps TDM staging traffic low.
    const int block = 512;
    const int grid  = 1024;

    np_sparse_interp_kernel<<<grid, block, TABLE_BYTES, stream>>>(
        positions, embeddings, block_tbl, out, npts);
}